// SetAbstraction_89283780149946
// MI455X (gfx1250) — compile-verified
//
#include <hip/hip_runtime.h>
#include <hip/hip_bf16.h>
#include <cstdint>
#include <cstddef>

typedef float v2f __attribute__((ext_vector_type(2)));
typedef float v8f __attribute__((ext_vector_type(8)));
typedef unsigned int u32x4 __attribute__((ext_vector_type(4)));
typedef int i32x8 __attribute__((ext_vector_type(8)));
typedef int i32x4 __attribute__((ext_vector_type(4)));

#define BB 16
#define NN 4096
#define CC 128
#define SS 1024
#define KK 32
#define MM (BB * SS * KK)   /* 524288 positions */
#define K1 132              /* stage-1 input channels padded 131 -> 132 */

// ---------------------------------------------------------------------------
// Farthest point sampling: one block per batch, 1024 serial steps.
// ---------------------------------------------------------------------------
__global__ void fps_kernel(const float* __restrict__ xyz,
                           int* __restrict__ fps_idx,
                           float* __restrict__ new_xyz) {
    const int b   = blockIdx.x;
    const int tid = threadIdx.x;                 // 256 threads
    const float* px = xyz + (size_t)b * NN * 3;
    __shared__ float rv[256];
    __shared__ int   ri[256];
    __shared__ int   s_far;

    float dist[NN / 256];
#pragma unroll
    for (int j = 0; j < NN / 256; ++j) dist[j] = 1e10f;
    if (tid == 0) s_far = 0;
    __syncthreads();

    for (int s = 0; s < SS; ++s) {
        const int far = s_far;
        if (tid == 0) {
            fps_idx[b * SS + s] = far;
            new_xyz[((size_t)b * SS + s) * 3 + 0] = px[far * 3 + 0];
            new_xyz[((size_t)b * SS + s) * 3 + 1] = px[far * 3 + 1];
            new_xyz[((size_t)b * SS + s) * 3 + 2] = px[far * 3 + 2];
        }
        const float cx = px[far * 3 + 0];
        const float cy = px[far * 3 + 1];
        const float cz = px[far * 3 + 2];
        float best = -1.0f; int bi = 0;
#pragma unroll
        for (int j = 0; j < NN / 256; ++j) {
            const int i = tid + j * 256;
            const float dx = px[i * 3 + 0] - cx;
            const float dy = px[i * 3 + 1] - cy;
            const float dz = px[i * 3 + 2] - cz;
            const float d  = dx * dx + dy * dy + dz * dz;
            const float nd = fminf(dist[j], d);
            dist[j] = nd;
            if (nd > best) { best = nd; bi = i; }
        }
        rv[tid] = best; ri[tid] = bi;
        __syncthreads();
        for (int off = 128; off; off >>= 1) {
            if (tid < off) {
                const float ov = rv[tid + off]; const int oi = ri[tid + off];
                if (ov > rv[tid] || (ov == rv[tid] && oi < ri[tid])) {
                    rv[tid] = ov; ri[tid] = oi;
                }
            }
            __syncthreads();
        }
        if (tid == 0) s_far = ri[0];
        __syncthreads();
    }
}

// ---------------------------------------------------------------------------
// kNN: one wave per query; dists in LDS; 32 wave32 butterfly min-selections.
// ---------------------------------------------------------------------------
__global__ void knn_kernel(const float* __restrict__ xyz,
                           const float* __restrict__ new_xyz,
                           int* __restrict__ knn_idx) {
    __shared__ float sd[2][NN];                  // 32 KB
    const int tid  = threadIdx.x;                // 64 threads = 2 waves
    const int wave = tid >> 5;
    const int lane = tid & 31;
    const int q    = blockIdx.x * 2 + wave;      // [0, B*S)
    const int b    = q >> 10;
    const float* px = xyz + (size_t)b * NN * 3;
    const float qx = new_xyz[(size_t)q * 3 + 0];
    const float qy = new_xyz[(size_t)q * 3 + 1];
    const float qz = new_xyz[(size_t)q * 3 + 2];

    for (int i = lane; i < NN; i += 32) {
        const float dx = px[i * 3 + 0] - qx;
        const float dy = px[i * 3 + 1] - qy;
        const float dz = px[i * 3 + 2] - qz;
        sd[wave][i] = dx * dx + dy * dy + dz * dz;
    }
    __syncthreads();

    for (int r = 0; r < KK; ++r) {
        float v = 3.0e38f; int vi = NN;
        for (int i = lane; i < NN; i += 32) {
            const float d = sd[wave][i];
            if (d < v || (d == v && i < vi)) { v = d; vi = i; }
        }
        for (int off = 16; off; off >>= 1) {
            const float ov = __shfl_xor(v, off, 32);
            const int   oi = __shfl_xor(vi, off, 32);
            if (ov < v || (ov == v && oi < vi)) { v = ov; vi = oi; }
        }
        if (lane == 0) {
            knn_idx[(size_t)q * KK + r] = vi;
            sd[wave][vi] = 3.0e38f;
        }
        __syncthreads();
    }
}

// ---------------------------------------------------------------------------
// Gather: X0 [M, 132] = concat(grouped_xyz - center, grouped feats, pad 0).
// ---------------------------------------------------------------------------
__global__ void gather_kernel(const float* __restrict__ xyz,
                              const float* __restrict__ feats,
                              const float* __restrict__ new_xyz,
                              const int* __restrict__ knn_idx,
                              float* __restrict__ X0) {
    const size_t t = (size_t)blockIdx.x * blockDim.x + threadIdx.x;
    if (t >= (size_t)MM * K1) return;
    const int    c  = (int)(t % K1);
    const size_t m  = t / K1;
    const int    kk = (int)(m % KK);
    const size_t bs = m / KK;                    // b*S + s
    const int    b  = (int)(bs >> 10);
    const int    j  = knn_idx[bs * KK + kk];
    float v;
    if (c < 3)            v = xyz[((size_t)b * NN + j) * 3 + c] - new_xyz[bs * 3 + c];
    else if (c < 3 + CC)  v = feats[((size_t)b * CC + (c - 3)) * NN + j];
    else                  v = 0.0f;
    X0[m * K1 + c] = v;
}

// Pad W1 [128,131] -> [128,132] (zero last column).
__global__ void wpad_kernel(const float* __restrict__ w1, float* __restrict__ w1p) {
    const int t = blockIdx.x * blockDim.x + threadIdx.x;
    if (t >= 128 * K1) return;
    const int o = t / K1, c = t % K1;
    w1p[t] = (c < 131) ? w1[o * 131 + c] : 0.0f;
}

// ---------------------------------------------------------------------------
// Issue a TDM copy of W [O*KD floats, contiguous] into LDS (one wave issues,
// all waves barrier after). D# per CDNA5 ISA:
//   group0: count=1 | lds_addr | global_addr[56:0] | type=2
//   group1: data_size=4B, tensor_dim0=tile_dim0=O*KD, dims1/2=1 row
// This toolchain's builtin takes 6 args (g0, g1, g2, g3, g4, cpol).
// ---------------------------------------------------------------------------
__device__ __forceinline__ void tdm_load_w(const float* W, float* sW, int nelem) {
    if (threadIdx.x == 0) {
        const unsigned long long ga = (unsigned long long)(uintptr_t)W;
        const unsigned lds_off = (unsigned)(uintptr_t)sW;   // addr[31:0] = LDS offset
        u32x4 g0;
        g0[0] = 1u;                                  // count=1, user descriptor
        g0[1] = lds_off;                             // lds_addr (bytes)
        g0[2] = (unsigned)(ga & 0xffffffffull);      // global_addr[31:0]
        g0[3] = (unsigned)((ga >> 32) & 0x1ffffffull) | (2u << 30);  // [56:32] | type=2
        i32x8 g1;
        g1[0] = (int)(2u << 16);                     // data_size = 4 bytes
        g1[1] = (int)((unsigned)(nelem & 0xffff) << 16);          // tensor_dim0[15:0]
        g1[2] = (int)(((unsigned)nelem >> 16) | (1u << 16));      // dim0[31:16] | dim1=1
        g1[3] = (int)((unsigned)(nelem & 0xffff) << 16);          // tile_dim0[15:0]
        g1[4] = (int)(1u);                           // tile_dim1=1, tile_dim2=0
        g1[5] = (int)(unsigned)nelem;                // tensor_dim0_stride[31:0]
        g1[6] = 0;                                   // stride0[47:32] | stride1[15:0]
        g1[7] = 0;                                   // stride1[47:16]
        i32x4 g2 = {0, 0, 0, 0};
        i32x4 g3 = {0, 0, 0, 0};
        i32x8 g4 = {0, 0, 0, 0, 0, 0, 0, 0};
        __builtin_amdgcn_tensor_load_to_lds(g0, g1, g2, g3, g4, 0);
        __builtin_amdgcn_s_wait_tensorcnt(0);
    }
    __syncthreads();
}

// ---------------------------------------------------------------------------
// GEMM via V_WMMA_F32_16X16X4_F32, compile-time shapes: Y[M,O] = act(X)*W^T.
// W staged in LDS by the Tensor Data Mover; 16xKD A-strip staged in LDS
// (shared by the 8 waves) with optional fused BN+ReLU; branch-free unrolled
// k-loop of ds_load_b64 + wmma.
// ---------------------------------------------------------------------------
template <int KD, int O, int FUSE>
__global__ void gemm_kernel(const float* __restrict__ X,
                            const float* __restrict__ W,
                            float* __restrict__ Y,
                            const float* __restrict__ aArr,
                            const float* __restrict__ bArr) {
    __shared__ __align__(16) float sW[O * KD];
    __shared__ __align__(16) float sA[16 * KD];
    __shared__ float sa[KD];
    __shared__ float sb[KD];
    const int tid  = threadIdx.x;                // 256 threads = 8 waves
    const int wave = tid >> 5;
    const int lane = tid & 31;

    if (FUSE) {
        for (int c = tid; c < KD; c += 256) { sa[c] = aArr[c]; sb[c] = bArr[c]; }
    }
    tdm_load_w(W, sW, O * KD);                   // includes block barrier

    const int nstrip = MM / 16;
    const int mrow   = lane & 15;
    const int koff   = (lane >> 4) * 2;
    const int ncol   = lane & 15;
    const int n0     = wave * 16;

    for (int strip = blockIdx.x; strip < nstrip; strip += gridDim.x) {
        const float* Xs = X + (size_t)strip * 16 * KD;
        const int nextStrip = strip + gridDim.x;
        if (nextStrip < nstrip)
            __builtin_prefetch(X + (size_t)nextStrip * 16 * KD + tid * 16, 0, 0);

        // cooperative staging of the 16 x KD A-strip (float4, fused BN+ReLU)
        for (int t = tid; t < (16 * KD) / 4; t += 256) {
            float4 v = ((const float4*)Xs)[t];
            if (FUSE) {
                const int c = (t * 4) % KD;
                v.x = fmaxf(sa[c + 0] * v.x + sb[c + 0], 0.0f);
                v.y = fmaxf(sa[c + 1] * v.y + sb[c + 1], 0.0f);
                v.z = fmaxf(sa[c + 2] * v.z + sb[c + 2], 0.0f);
                v.w = fmaxf(sa[c + 3] * v.w + sb[c + 3], 0.0f);
            }
            ((float4*)sA)[t] = v;
        }
        __syncthreads();

        v8f acc0 = {};
        v8f acc1 = {};
#pragma unroll
        for (int k0 = 0; k0 < KD; k0 += 4) {
            const v2f a  = *(const v2f*)&sA[mrow * KD + k0 + koff];
            const v2f b0 = *(const v2f*)&sW[(n0 + ncol) * KD + k0 + koff];
            acc0 = __builtin_amdgcn_wmma_f32_16x16x4_f32(
                false, a, false, b0, (short)0, acc0, false, false);
            if (O == 256) {
                const v2f b1 = *(const v2f*)&sW[(n0 + 128 + ncol) * KD + k0 + koff];
                acc1 = __builtin_amdgcn_wmma_f32_16x16x4_f32(
                    false, a, false, b1, (short)0, acc1, false, false);
            }
        }

        const int rbase = strip * 16 + 8 * (lane >> 4);
#pragma unroll
        for (int r = 0; r < 8; ++r) {
            Y[(size_t)(rbase + r) * O + n0 + ncol] = acc0[r];
            if (O == 256)
                Y[(size_t)(rbase + r) * O + n0 + 128 + ncol] = acc1[r];
        }
        __syncthreads();
    }
}

// Per-channel sum / sum-of-squares over M (training-mode BN stats).
__global__ void reduce_kernel(const float* __restrict__ Y, int O,
                              float* __restrict__ sums, float* __restrict__ sumsq) {
    const int c = blockIdx.x * 64 + threadIdx.x;
    float s = 0.0f, ss = 0.0f;
    for (size_t m = blockIdx.y; m < (size_t)MM; m += gridDim.y) {
        const float v = Y[m * O + c];
        s += v; ss += v * v;
    }
    atomicAdd(&sums[c], s);
    atomicAdd(&sumsq[c], ss);
}

__global__ void zero_stats(float* __restrict__ sums, float* __restrict__ sumsq) {
    sums[threadIdx.x]  = 0.0f;
    sumsq[threadIdx.x] = 0.0f;
}

// Fold mean/var/gamma/beta into y = a*x + b (relu applied by the consumer).
__global__ void finalize_kernel(const float* __restrict__ sums,
                                const float* __restrict__ sumsq,
                                const float* __restrict__ g,
                                const float* __restrict__ bt,
                                float* __restrict__ aArr, float* __restrict__ bArr,
                                int O) {
    const int c = threadIdx.x;
    if (c >= O) return;
    const float inv  = 1.0f / (float)MM;
    const float mean = sums[c] * inv;
    const float var  = sumsq[c] * inv - mean * mean;
    const float a    = g[c] * rsqrtf(var + 1e-5f);
    aArr[c] = a;
    bArr[c] = bt[c] - mean * a;
}

// Fused BN+ReLU+max over K; output layout [B, 256, S].
__global__ void maxpool_kernel(const float* __restrict__ Y3,
                               const float* __restrict__ aArr,
                               const float* __restrict__ bArr,
                               float* __restrict__ outF) {
    const int s = blockIdx.x, b = blockIdx.y, o = threadIdx.x;   // 256 channels
    const size_t base = (((size_t)b * SS + s) * KK) * 256;
    const float a = aArr[o], bb = bArr[o];
    float vmax = 0.0f;                        // relu output is >= 0
    for (int kk = 0; kk < KK; ++kk) {
        float v = Y3[base + (size_t)kk * 256 + o];
        v = fmaxf(a * v + bb, 0.0f);
        vmax = fmaxf(vmax, v);
    }
    outF[((size_t)b * 256 + o) * SS + s] = vmax;
}

// ---------------------------------------------------------------------------
extern "C" void kernel_launch(void* const* d_in, const int* in_sizes, int n_in,
                              void* d_out, int out_size, void* d_ws, size_t ws_size,
                              hipStream_t stream) {
    const float* xyz   = (const float*)d_in[0];
    const float* feats = (const float*)d_in[1];
    const float* w1    = (const float*)d_in[2];
    const float* g1    = (const float*)d_in[3];
    const float* b1    = (const float*)d_in[4];
    const float* w2    = (const float*)d_in[5];
    const float* g2    = (const float*)d_in[6];
    const float* b2    = (const float*)d_in[7];
    const float* w3    = (const float*)d_in[8];
    const float* g3    = (const float*)d_in[9];
    const float* b3    = (const float*)d_in[10];
    (void)in_sizes; (void)n_in; (void)out_size; (void)ws_size;

    float* out_xyz  = (float*)d_out;                       // [B,S,3]
    float* out_feat = (float*)d_out + (size_t)BB * SS * 3; // [B,256,S]

    char* ws = (char*)d_ws;
    size_t off = 0;
    int*   fps_idx = (int*)(ws + off);   off += (size_t)BB * SS * 4;
    int*   knn_idx = (int*)(ws + off);   off += (size_t)BB * SS * KK * 4;
    float* w1p     = (float*)(ws + off); off += (size_t)128 * K1 * 4;
    float* sums    = (float*)(ws + off); off += 256 * 4;
    float* sumsq   = (float*)(ws + off); off += 256 * 4;
    float* aArr    = (float*)(ws + off); off += 256 * 4;
    float* bArr    = (float*)(ws + off); off += 256 * 4;
    off = (off + 4095) & ~(size_t)4095;
    float* X0      = (float*)(ws + off); off += (size_t)MM * K1 * 4;   // reused as Y2
    float* Y1      = (float*)(ws + off); off += (size_t)MM * 128 * 4;
    float* Y3      = (float*)(ws + off); off += (size_t)MM * 256 * 4;
    float* Y2      = X0;   // stage-2 output overwrites gathered input

    // 1) FPS + new_xyz
    fps_kernel<<<BB, 256, 0, stream>>>(xyz, fps_idx, out_xyz);
    // 2) kNN
    knn_kernel<<<(BB * SS) / 2, 64, 0, stream>>>(xyz, out_xyz, knn_idx);
    // 3) gather + concat (padded to 132 channels)
    {
        const size_t tot = (size_t)MM * K1;
        gather_kernel<<<(unsigned)((tot + 255) / 256), 256, 0, stream>>>(
            xyz, feats, out_xyz, knn_idx, X0);
    }
    wpad_kernel<<<(128 * K1 + 255) / 256, 256, 0, stream>>>(w1, w1p);

    // 4) conv1 (raw GEMM) -> stats -> fold
    zero_stats<<<1, 256, 0, stream>>>(sums, sumsq);
    gemm_kernel<K1, 128, 0><<<4096, 256, 0, stream>>>(X0, w1p, Y1, aArr, bArr);
    reduce_kernel<<<dim3(2, 1024), 64, 0, stream>>>(Y1, 128, sums, sumsq);
    finalize_kernel<<<1, 256, 0, stream>>>(sums, sumsq, g1, b1, aArr, bArr, 128);

    // 5) conv2 (BN1+ReLU fused into A staging) -> stats -> fold
    zero_stats<<<1, 256, 0, stream>>>(sums, sumsq);
    gemm_kernel<128, 128, 1><<<4096, 256, 0, stream>>>(Y1, w2, Y2, aArr, bArr);
    reduce_kernel<<<dim3(2, 1024), 64, 0, stream>>>(Y2, 128, sums, sumsq);
    finalize_kernel<<<1, 256, 0, stream>>>(sums, sumsq, g2, b2, aArr, bArr, 128);

    // 6) conv3 (BN2+ReLU fused), O=256 -> stats -> fold
    zero_stats<<<1, 256, 0, stream>>>(sums, sumsq);
    gemm_kernel<128, 256, 1><<<4096, 256, 0, stream>>>(Y2, w3, Y3, aArr, bArr);
    reduce_kernel<<<dim3(4, 1024), 64, 0, stream>>>(Y3, 256, sums, sumsq);
    finalize_kernel<<<1, 256, 0, stream>>>(sums, sumsq, g3, b3, aArr, bArr, 256);

    // 7) BN3+ReLU+max over K
    maxpool_kernel<<<dim3(SS, BB), 256, 0, stream>>>(Y3, aArr, bArr, out_feat);
}